// SAGE_40776419508824
// MI455X (gfx1250) — compile-verified
//
#include <hip/hip_runtime.h>

#define N_NODES 100000

typedef __attribute__((ext_vector_type(2))) float v2f;
typedef __attribute__((ext_vector_type(8))) float v8f;

// ---------------------------------------------------------------------------
// Zero a region of workspace (grid-stride).
// ---------------------------------------------------------------------------
__global__ void sage_zero_kernel(float* __restrict__ p, long long n) {
    long long i = (long long)blockIdx.x * blockDim.x + threadIdx.x;
    long long stride = (long long)gridDim.x * blockDim.x;
    for (; i < n; i += stride) p[i] = 0.0f;
}

// ---------------------------------------------------------------------------
// Weight transpose: out[n*K + k] = in[k*N + n].  (tiny: <=32KB per matrix)
// Makes the WMMA B-fragment a contiguous float2 -> one global_load_b64.
// ---------------------------------------------------------------------------
__global__ void sage_transpose_kernel(const float* __restrict__ in,
                                      float* __restrict__ out, int K, int N) {
    int idx = blockIdx.x * blockDim.x + threadIdx.x;
    if (idx >= K * N) return;
    int k = idx / N;
    int n = idx - k * N;
    out[(size_t)n * K + k] = in[idx];
}

// ---------------------------------------------------------------------------
// Edge scatter-add: agg[dst[e], :] += X[src[e], :], plus fused in-degree.
// D features per edge, 4 features per thread (float4 gather, 4 f32 atomics).
// ---------------------------------------------------------------------------
template <int D, bool DO_DEG>
__global__ void sage_scatter_kernel(const float* __restrict__ X,
                                    const int* __restrict__ src,
                                    const int* __restrict__ dst,
                                    float* __restrict__ agg,
                                    float* __restrict__ deg, int E) {
    constexpr int T = D / 4;  // threads per edge (16 or 32)
    long long tid = (long long)blockIdx.x * blockDim.x + threadIdx.x;
    int e = (int)(tid / T);
    if (e >= E) return;
    int f = ((int)(tid % T)) * 4;
    int s = src[e];
    int d = dst[e];
    if (DO_DEG && f == 0) atomicAdd(&deg[d], 1.0f);
    const float4 v = *(const float4*)(X + (size_t)s * D + f);
    float* p = agg + (size_t)d * D + f;
    atomicAdd(p + 0, v.x);
    atomicAdd(p + 1, v.y);
    atomicAdd(p + 2, v.z);
    atomicAdd(p + 3, v.w);
}

// ---------------------------------------------------------------------------
// Fused SAGE layer GEMM:
//   Out[m, n] = act( sum_k Xself[m,k]*WselfT[n,k]
//                  + sum_k (Agg[m,k]/max(deg[m],1))*WneighT[n,k] + bias[n] )
//
// One wave = one 16x16 output tile (V_WMMA_F32_16X16X4_F32).
// blockIdx.x -> 16-row slab; wave id -> 16-col tile.
//
// The two 16xKDIM A-tiles (contiguous slabs) are DMA'd into LDS once per
// block via global_load_async_to_lds_b128 (ASYNCcnt), then each wave feeds
// its A-fragments from LDS (ds_load_b64). B-fragments are contiguous
// float2 loads from the pre-transposed weights.
//
// Fragment layouts per CDNA5 ISA 7.12.2:
//   A (16x4 f32):  lanes 0-15 = rows, VGPR{0,1} = K{0,1}; lanes 16-31 = K{2,3}
//   B (4x16 f32):  lanes 0-15 = cols, VGPR{0,1} = K{0,1}; lanes 16-31 = K{2,3}
//   C/D (16x16):   lanes 0-15 = cols N=lane, VGPR v = row M=v (+8 hi lanes)
// ---------------------------------------------------------------------------
template <int KDIM, int NCOLS, bool RELU>
__global__ void sage_gemm_kernel(const float* __restrict__ Xself,
                                 const float* __restrict__ Agg,
                                 const float* __restrict__ deg,
                                 const float* __restrict__ WselfT,
                                 const float* __restrict__ WneighT,
                                 const float* __restrict__ bias,
                                 float* __restrict__ Out) {
    __shared__ float sA[16 * KDIM];  // Xself tile
    __shared__ float sG[16 * KDIM];  // Agg tile

    const int tid = threadIdx.x;
    constexpr int NTHREADS = NCOLS * 2;  // (NCOLS/16) waves * 32 lanes
    const int m0 = blockIdx.x << 4;

    // --- Async DMA stage: 16xKDIM contiguous slabs -> LDS ------------------
    {
        const float* gA = Xself + (size_t)m0 * KDIM;
        const float* gG = Agg + (size_t)m0 * KDIM;
#pragma unroll
        for (int i = tid * 4; i < 16 * KDIM; i += NTHREADS * 4) {
            unsigned lA = (unsigned)(size_t)(&sA[i]);  // low 32 bits = LDS addr
            unsigned lG = (unsigned)(size_t)(&sG[i]);
            asm volatile("global_load_async_to_lds_b128 %0, %1, off"
                         :: "v"(lA), "v"(gA + i) : "memory");
            asm volatile("global_load_async_to_lds_b128 %0, %1, off"
                         :: "v"(lG), "v"(gG + i) : "memory");
        }
        asm volatile("s_wait_asynccnt 0x0" ::: "memory");
    }
    __syncthreads();

    // --- WMMA main loops ---------------------------------------------------
    const int lane = tid & 31;
    const int wave = tid >> 5;
    const int n0 = wave << 4;
    const int r = lane & 15;            // A row / B-C-D column sub-index
    const int lcol = n0 + r;
    const int kh = (lane >> 4) << 1;    // K sub-offset: 0 (lo half) / 2 (hi)

    const float inv = 1.0f / fmaxf(deg[m0 + r], 1.0f);

    v8f acc = {};

    // Self term: Xself @ Wself
#pragma unroll
    for (int k0 = 0; k0 < KDIM; k0 += 4) {
        v2f a = *(const v2f*)(&sA[r * KDIM + k0 + kh]);
        v2f b = *(const v2f*)(WselfT + (size_t)lcol * KDIM + k0 + kh);
        acc = __builtin_amdgcn_wmma_f32_16x16x4_f32(
            /*neg_a=*/false, a, /*neg_b=*/false, b,
            /*c_mod=*/(short)0, acc, /*reuse_a=*/false, /*reuse_b=*/false);
    }

    // Neighbor term: (Agg * invdeg_row) @ Wneigh (row scale fused into A frag)
#pragma unroll
    for (int k0 = 0; k0 < KDIM; k0 += 4) {
        v2f a = *(const v2f*)(&sG[r * KDIM + k0 + kh]);
        a.x *= inv;
        a.y *= inv;
        v2f b = *(const v2f*)(WneighT + (size_t)lcol * KDIM + k0 + kh);
        acc = __builtin_amdgcn_wmma_f32_16x16x4_f32(
            false, a, false, b, (short)0, acc, false, false);
    }

    // --- Epilogue: bias (+ ReLU), store D tile -----------------------------
    const float bc = bias[lcol];
    const int mbase = m0 + ((lane >> 4) << 3);
#pragma unroll
    for (int v = 0; v < 8; ++v) {
        float val = acc[v] + bc;
        if (RELU) val = fmaxf(val, 0.0f);
        Out[(size_t)(mbase + v) * NCOLS + lcol] = val;
    }
}

// ---------------------------------------------------------------------------
// Launch: transpose weights + zero ws -> scatter1(+deg) -> GEMM1(relu)
//         -> scatter2 -> GEMM2
// ---------------------------------------------------------------------------
extern "C" void kernel_launch(void* const* d_in, const int* in_sizes, int n_in,
                              void* d_out, int out_size, void* d_ws, size_t ws_size,
                              hipStream_t stream) {
    const float* x        = (const float*)d_in[0];
    const int*   src      = (const int*)d_in[1];
    const int*   dst      = (const int*)d_in[2];
    const float* w_self1  = (const float*)d_in[3];
    const float* w_neigh1 = (const float*)d_in[4];
    const float* b1       = (const float*)d_in[5];
    const float* w_self2  = (const float*)d_in[6];
    const float* w_neigh2 = (const float*)d_in[7];
    const float* b2       = (const float*)d_in[8];
    const int E = in_sizes[1];

    float* ws   = (float*)d_ws;
    float* deg  = ws;                              // N
    float* agg1 = deg + N_NODES;                   // N * 64
    float* agg2 = agg1 + (size_t)N_NODES * 64;     // N * 128
    float* h    = agg2 + (size_t)N_NODES * 128;    // N * 128
    float* wts1 = h + (size_t)N_NODES * 128;       // 128 x 64  (w_self1^T)
    float* wtn1 = wts1 + 128 * 64;                 // 128 x 64  (w_neigh1^T)
    float* wts2 = wtn1 + 128 * 64;                 // 64 x 128  (w_self2^T)
    float* wtn2 = wts2 + 64 * 128;                 // 64 x 128  (w_neigh2^T)
    float* out  = (float*)d_out;                   // N * 64

    // Transpose the four weight matrices (independent, tiny).
    sage_transpose_kernel<<<(64 * 128 + 255) / 256, 256, 0, stream>>>(w_self1,  wts1, 64, 128);
    sage_transpose_kernel<<<(64 * 128 + 255) / 256, 256, 0, stream>>>(w_neigh1, wtn1, 64, 128);
    sage_transpose_kernel<<<(128 * 64 + 255) / 256, 256, 0, stream>>>(w_self2,  wts2, 128, 64);
    sage_transpose_kernel<<<(128 * 64 + 255) / 256, 256, 0, stream>>>(w_neigh2, wtn2, 128, 64);

    // Zero deg + agg1 + agg2 (one contiguous region at the front of ws).
    const long long nzero = (long long)N_NODES * (1 + 64 + 128);
    sage_zero_kernel<<<2048, 256, 0, stream>>>(ws, nzero);

    // Layer 1 aggregation (+ in-degree): agg1[dst] += x[src]
    {
        long long tot = (long long)E * 16;
        sage_scatter_kernel<64, true><<<(unsigned)((tot + 255) / 256), 256, 0, stream>>>(
            x, src, dst, agg1, deg, E);
    }

    // Layer 1 GEMM + ReLU: h = relu(x@Wself1 + (agg1/deg)@Wneigh1 + b1)
    sage_gemm_kernel<64, 128, true><<<N_NODES / 16, 256, 0, stream>>>(
        x, agg1, deg, wts1, wtn1, b1, h);

    // Layer 2 aggregation: agg2[dst] += h[src]
    {
        long long tot = (long long)E * 32;
        sage_scatter_kernel<128, false><<<(unsigned)((tot + 255) / 256), 256, 0, stream>>>(
            h, src, dst, agg2, deg, E);
    }

    // Layer 2 GEMM: out = h@Wself2 + (agg2/deg)@Wneigh2 + b2
    sage_gemm_kernel<128, 64, false><<<N_NODES / 16, 128, 0, stream>>>(
        h, agg2, deg, wts2, wtn2, b2, out);
}